// get_loss_pre_4973572129196
// MI455X (gfx1250) — compile-verified
//
#include <hip/hip_runtime.h>
#include <math.h>

// Problem constants (from reference setup_inputs)
#define BB 16
#define NN 4096
#define MM 1024
#define M_TILE 128          // columns per workgroup (8 waves x 16 cols)
#define N_CHUNK 128         // rows staged in LDS per outer iteration

#define EPS_BITS 0x2B8CBCCC // bit pattern of 1e-12f
#define INF_BITS 0x7F800000 // bit pattern of +inf

typedef float v2f __attribute__((ext_vector_type(2)));
typedef float v8f __attribute__((ext_vector_type(8)));

// ---------------------------------------------------------------------------
// Butterfly min across each 16-lane half (one DPP16 row on wave32) in the
// UNSIGNED-INT domain: clamped d^2 bit patterns are positive, so u32 order ==
// float order -> fused v_min_u32_dpp, no canonicalization. Stages pair
// disjoint sets => valid reduction for min.
// ---------------------------------------------------------------------------
__device__ __forceinline__ unsigned int dpp_min16_u32(unsigned int x) {
    unsigned int t;
    t = (unsigned int)__builtin_amdgcn_update_dpp(0, (int)x, 0xB1, 0xF, 0xF, true);  // quad_perm(1,0,3,2)
    x = (t < x) ? t : x;
    t = (unsigned int)__builtin_amdgcn_update_dpp(0, (int)x, 0x4E, 0xF, 0xF, true);  // quad_perm(2,3,0,1)
    x = (t < x) ? t : x;
    t = (unsigned int)__builtin_amdgcn_update_dpp(0, (int)x, 0x141, 0xF, 0xF, true); // row_half_mirror
    x = (t < x) ? t : x;
    t = (unsigned int)__builtin_amdgcn_update_dpp(0, (int)x, 0x140, 0xF, 0xF, true); // row_mirror
    x = (t < x) ? t : x;
    return x;
}

// ---------------------------------------------------------------------------
// Kernel 1: init workspace. rowmin[b*N+n] = +inf (bit pattern), partials = 0.
// ---------------------------------------------------------------------------
__global__ void init_kernel(unsigned int* __restrict__ rowmin,
                            float* __restrict__ partials,
                            int nRow, int nPart) {
    int i = blockIdx.x * blockDim.x + threadIdx.x;
    if (i < nRow)  rowmin[i]   = (unsigned int)INF_BITS;
    if (i < nPart) partials[i] = 0.0f;
}

// ---------------------------------------------------------------------------
// Kernel 2: main. grid = BB * (MM / M_TILE) workgroups of 256 threads.
// d^2 = |p|^2 - 2 p.q + |q|^2 as one V_WMMA_F32_16X16X4_F32 per 16x16 tile:
// A = [-2x,-2y,-2z,1], B = [qx,qy,qz,|q|^2], C preloaded with |p|^2.
// All selection/reduction happens on d^2 *bit patterns* in the integer domain
// (sqrt is monotonic; positive-float bits order as ints); the EPS clamp is a
// single v_max_i32 since negative-float bits are negative int32s.
// sqrt is applied only to the final per-column / per-row minima.
// ---------------------------------------------------------------------------
__global__ void __launch_bounds__(256)
main_kernel(const float* __restrict__ shape,    // (B,N,6): xyz + normal
            const float* __restrict__ skelx,    // (B,M,3)
            const float* __restrict__ skeln,    // (B,M,3)
            unsigned int* __restrict__ rowmin,  // (B*N) d^2 float-bits, atomicMin
            float* __restrict__ partials) {     // (2 * gridDim.x)
    __shared__ float4 sP[N_CHUNK];              // x,y,z,|p|^2 for current chunk
    __shared__ float  sCd[8], sNl[8];

    const int tid  = threadIdx.x;
    const int wave = tid >> 5;
    const int lane = tid & 31;
    const int col  = lane & 15;                 // column within wave tile / A row
    const int hi   = lane >> 4;                 // 0: K=0,1 half   1: K=2,3 half

    const int b         = blockIdx.x >> 3;                 // MM/M_TILE == 8
    const int mTileBase = (blockIdx.x & 7) * M_TILE;
    const int m         = mTileBase + wave * 16 + col;     // this lane's column

    // --- Loop-invariant B operand: [qx,qy,qz,|q|^2] striped per ISA layout ---
    const float* qp = skelx + ((size_t)b * MM + m) * 3;
    const float qx = qp[0], qy = qp[1], qz = qp[2];
    const float qn2 = qx * qx + qy * qy + qz * qz;
    v2f bmat;
    bmat.x = hi ? qz  : qx;     // K=2 : K=0
    bmat.y = hi ? qn2 : qy;     // K=3 : K=1

    // Per-column running top-2 (smallest d^2 bit patterns) with row indices
    int d1i = INF_BITS, d2i = INF_BITS;
    int i1 = 0, i2 = 0;

    for (int nc = 0; nc < NN; nc += N_CHUNK) {
        __syncthreads();
        if (tid < N_CHUNK) {                    // stage 128 rows cooperatively
            const float* sp = shape + ((size_t)b * NN + nc + tid) * 6;
            float px = sp[0], py = sp[1], pz = sp[2];
            sP[tid] = make_float4(px, py, pz, px * px + py * py + pz * pz);
            if (nc + N_CHUNK < NN)              // global_prefetch next chunk
                __builtin_prefetch(sp + (size_t)N_CHUNK * 6, 0, 3);
        }
        __syncthreads();

        #pragma unroll
        for (int sub = 0; sub < N_CHUNK / 16; ++sub) {
            const int rowBase = sub * 16;

            // A operand: lane supplies row (lane&15); hi half supplies K=2,3.
            float4 p = sP[rowBase + col];
            v2f amat;
            amat.x = hi ? (-2.0f * p.z) : (-2.0f * p.x);   // K=2 : K=0
            amat.y = hi ? 1.0f          : (-2.0f * p.y);   // K=3 : K=1

            // C operand: |p|^2 for row (j + 8*hi)  -> D = |p|^2 - 2p.q + |q|^2
            v8f c;
            #pragma unroll
            for (int j = 0; j < 8; ++j) c[j] = sP[rowBase + j + 8 * hi].w;

            v8f d2t = __builtin_amdgcn_wmma_f32_16x16x4_f32(
                false, amat, false, bmat, (short)0, c, false, false);

            // Row-min accumulator: lane with col==j captures the (replicated)
            // butterfly result for row rowBase+j+8*hi; one atomic per tile.
            unsigned int accMin = (unsigned int)INF_BITS;

            #pragma unroll
            for (int j = 0; j < 8; ++j) {
                const int n = nc + rowBase + j + 8 * hi;       // this row index
                // EPS clamp in one v_max_i32: negative-float bits are negative
                // ints; positive floats order identically as int32.
                int db = __float_as_int(d2t[j]);
                db = (db < EPS_BITS) ? EPS_BITS : db;

                // branch-free per-column top-2 update (pure cndmask, int domain)
                const bool b1 = db < d1i;
                const bool b2 = db < d2i;
                const int nd2 = b1 ? d1i : (b2 ? db : d2i);
                const int ni2 = b1 ? i1  : (b2 ? n  : i2);
                d2i = nd2;  i2 = ni2;
                i1  = b1 ? n  : i1;
                d1i = b1 ? db : d1i;

                // per-row min across 16 lanes of this half (u32 DPP butterfly),
                // captured into the lane that owns row j of this half
                const unsigned int rm = dpp_min16_u32((unsigned int)db);
                accMin = (col == j) ? rm : accMin;
            }

            // Single predicated atomic per 16x16 tile: 16 active lanes
            // (cols 0-7 of each half), each addressing its own row.
            if (col < 8)
                atomicMin(&rowmin[b * NN + nc + rowBase + col + 8 * hi], accMin);
        }
    }

    // --- Merge top-2 between lane L (rows 0-7 of tiles) and lane L+16 ---
    int od1 = __shfl_xor(d1i, 16);
    int oi1 = __shfl_xor(i1, 16);
    int od2 = __shfl_xor(d2i, 16);
    int oi2 = __shfl_xor(i2, 16);
    if (od1 < d1i)      { d2i = d1i; i2 = i1; d1i = od1; i1 = oi1; }
    else if (od1 < d2i) { d2i = od1; i2 = oi1; }
    if (od2 < d1i)      { d2i = d1i; i2 = i1; d1i = od2; i1 = oi2; }
    else if (od2 < d2i) { d2i = od2; i2 = oi2; }

    // --- Column contributions: cd2 = sqrt(min d^2); normal loss = mean |dot| ---
    float cd2c = 0.0f, nlc = 0.0f;
    if (!hi) {
        cd2c = __builtin_amdgcn_sqrtf(__int_as_float(d1i)); // one sqrt / column
        const float* nr = skeln + ((size_t)b * MM + m) * 3;
        const float nx = nr[0], ny = nr[1], nz = nr[2];
        const float* f1 = shape + ((size_t)b * NN + i1) * 6 + 3;
        const float* f2 = shape + ((size_t)b * NN + i2) * 6 + 3;
        const float dot1 = nx * f1[0] + ny * f1[1] + nz * f1[2];
        const float dot2 = nx * f2[0] + ny * f2[1] + nz * f2[2];
        nlc = 0.5f * (fabsf(dot1) + fabsf(dot2));
    }

    // --- Deterministic per-workgroup reduction of the two partial sums ---
    #pragma unroll
    for (int mask = 16; mask >= 1; mask >>= 1) {
        cd2c += __shfl_xor(cd2c, mask);
        nlc  += __shfl_xor(nlc, mask);
    }
    if (lane == 0) { sCd[wave] = cd2c; sNl[wave] = nlc; }
    __syncthreads();
    if (tid == 0) {
        float a = 0.0f, bsum = 0.0f;
        #pragma unroll
        for (int w = 0; w < 8; ++w) { a += sCd[w]; bsum += sNl[w]; }
        partials[2 * blockIdx.x]     = a;
        partials[2 * blockIdx.x + 1] = bsum;
    }
}

// ---------------------------------------------------------------------------
// Kernel 3: finalize. rowmin holds min d^2 bit patterns: sqrt here (65K ops),
// then fixed-shape deterministic reduction -> scalar loss.
// ---------------------------------------------------------------------------
__global__ void finalize_kernel(const unsigned int* __restrict__ rowmin,
                                const float* __restrict__ partials,
                                int numWG, float* __restrict__ out) {
    __shared__ float s1[256], s2[256], s3[256];
    const int t = threadIdx.x;
    float c1 = 0.0f;
    for (int i = t; i < BB * NN; i += 256)
        c1 += __builtin_amdgcn_sqrtf(__uint_as_float(rowmin[i]));
    float c2 = 0.0f, nl = 0.0f;
    for (int i = t; i < numWG; i += 256) {
        c2 += partials[2 * i];
        nl += partials[2 * i + 1];
    }
    s1[t] = c1; s2[t] = c2; s3[t] = nl;
    __syncthreads();
    for (int off = 128; off > 0; off >>= 1) {
        if (t < off) { s1[t] += s1[t + off]; s2[t] += s2[t + off]; s3[t] += s3[t + off]; }
        __syncthreads();
    }
    if (t == 0)
        out[0] = 1.0e-4f * (s1[0] + s2[0]) + 1.0e-3f * (s3[0] / (float)BB);
}

// ---------------------------------------------------------------------------
extern "C" void kernel_launch(void* const* d_in, const int* in_sizes, int n_in,
                              void* d_out, int out_size, void* d_ws, size_t ws_size,
                              hipStream_t stream) {
    const float* shape = (const float*)d_in[0];   // (16,4096,6)
    const float* skelx = (const float*)d_in[1];   // (16,1024,3)
    const float* skeln = (const float*)d_in[2];   // (16,1024,3)
    float* out = (float*)d_out;

    unsigned int* rowmin = (unsigned int*)d_ws;                          // B*N
    float* partials = (float*)((char*)d_ws +
                               (size_t)BB * NN * sizeof(unsigned int));  // 2*numWG

    const int numWG = BB * (MM / M_TILE);          // 16 * 8 = 128
    const int nRow  = BB * NN;                     // 65536

    init_kernel<<<(nRow + 255) / 256, 256, 0, stream>>>(rowmin, partials,
                                                        nRow, 2 * numWG);
    main_kernel<<<numWG, 256, 0, stream>>>(shape, skelx, skeln, rowmin, partials);
    finalize_kernel<<<1, 256, 0, stream>>>(rowmin, partials, numWG, out);
}